// Likelyloss_46445776339738
// MI455X (gfx1250) — compile-verified
//
#include <hip/hip_runtime.h>

typedef __attribute__((ext_vector_type(2))) float v2f;
typedef __attribute__((ext_vector_type(8))) float v8f;

#define BATCH 16
#define NPIX (512*512)
#define EPSV 1e-10f
#define TWO_PI 6.283185307179586f

// ws float layout
#define WS_STATS 0      // 16 * 40  (stats[b][row 0..4][col 0..6], stride 8)
#define WS_MU    640    // 16 * 12
#define WS_H     832    // 16 * 12
#define WS_PK    1024   // 16 * 4
#define WS_NUM   1088   // 16
#define WS_TOTAL 1104

__global__ void k_zero(float* __restrict__ ws) {
    int i = blockIdx.x * blockDim.x + threadIdx.x;
    if (i < WS_TOTAL) ws[i] = 0.0f;
}

// -------- Phase 1: WMMA-accumulated masked moment reductions --------
// A rows (M): 0..3 = pred_k*mask, 4 = mask, 5..15 = 0
// B cols (N): 0 = 1, 1..3 = inp_m, 4..6 = inp_m^2, 7..15 = 0
// K dim = pixels (4 per WMMA, 8 WMMAs per 32-pixel wave tile)
__global__ void k_phase1(const float* __restrict__ pred,
                         const float* __restrict__ inp,
                         const int*   __restrict__ heart,
                         float* __restrict__ stats) {
    __shared__ float Als[8][16][32];   // [wave][channel(A row)][pixel]
    __shared__ float Bls[8][16][32];   // [wave][channel(B col)][pixel]

    const int b    = blockIdx.y;
    const int tid  = threadIdx.x;
    const int wave = tid >> 5;
    const int lane = tid & 31;
    const int chS  = lane & 15;        // A-row / B-col selected by this lane
    const int sel  = lane >> 4;        // 0: K={0,1}, 1: K={2,3} within chunk

    // zero LDS once so unused rows/cols contribute 0 forever
    for (int i = tid; i < 8*16*32; i += blockDim.x) {
        (&Als[0][0][0])[i] = 0.0f;
        (&Bls[0][0][0])[i] = 0.0f;
    }
    __syncthreads();

    const float* pb = pred  + (size_t)b*5*NPIX + NPIX;  // channels 1..4
    const float* ib = inp   + (size_t)b*3*NPIX;
    const int*   hb = heart + (size_t)b*NPIX;

    v8f acc = {};

    for (int tile = blockIdx.x; tile < NPIX/256; tile += gridDim.x) {
        const int pix = tile*256 + wave*32 + lane;
        const float m  = (hb[pix] == 1) ? 1.0f : 0.0f;
        const float p0 = pb[0*NPIX + pix];
        const float p1 = pb[1*NPIX + pix];
        const float p2 = pb[2*NPIX + pix];
        const float p3 = pb[3*NPIX + pix];
        const float x0 = ib[0*NPIX + pix];
        const float x1 = ib[1*NPIX + pix];
        const float x2 = ib[2*NPIX + pix];

        Als[wave][0][lane] = p0*m;
        Als[wave][1][lane] = p1*m;
        Als[wave][2][lane] = p2*m;
        Als[wave][3][lane] = p3*m;
        Als[wave][4][lane] = m;
        Bls[wave][0][lane] = 1.0f;
        Bls[wave][1][lane] = x0;
        Bls[wave][2][lane] = x1;
        Bls[wave][3][lane] = x2;
        Bls[wave][4][lane] = x0*x0;
        Bls[wave][5][lane] = x1*x1;
        Bls[wave][6][lane] = x2*x2;

        asm volatile("" ::: "memory");  // keep LDS stores before fragment loads

#pragma unroll
        for (int c = 0; c < 8; ++c) {
            const int k0 = c*4 + sel*2;
            v2f a  = *(const v2f*)&Als[wave][chS][k0];
            v2f bb = *(const v2f*)&Bls[wave][chS][k0];
            acc = __builtin_amdgcn_wmma_f32_16x16x4_f32(
                false, a, false, bb, (short)0, acc, false, false);
        }
        asm volatile("" ::: "memory");
    }

    // C layout: VGPR r, lanes 0..15 -> C[r][N=lane]; rows 0..4 / cols 0..6 live
    if (lane < 7) {
        float* sb = stats + b*40;
        atomicAdd(&sb[0*8 + lane], acc[0]);
        atomicAdd(&sb[1*8 + lane], acc[1]);
        atomicAdd(&sb[2*8 + lane], acc[2]);
        atomicAdd(&sb[3*8 + lane], acc[3]);
        atomicAdd(&sb[4*8 + lane], acc[4]);
    }
}

// -------- Phase 2: per-(b,k,m) Gaussian parameters --------
__global__ void k_params(const float* __restrict__ stats,
                         float* __restrict__ mu, float* __restrict__ h,
                         float* __restrict__ Pk) {
    int t = threadIdx.x;              // t = b*4 + k, 64 threads
    if (t >= BATCH*4) return;
    const float* sb = stats + (t >> 2)*40 + (t & 3)*8;
    const float sw    = sb[0];
    const float denom = sw + EPSV;
    float P = 1.0f;
#pragma unroll
    for (int m = 0; m < 3; ++m) {
        const float S1  = sb[1 + m];
        const float S2  = sb[4 + m];
        const float muv = S1 / denom;
        const float var = (S2 - 2.0f*muv*S1 + muv*muv*sw) / denom + EPSV;
        mu[t*3 + m] = muv;
        h [t*3 + m] = 0.5f / var;
        P *= rsqrtf(TWO_PI * var);
    }
    Pk[t] = P;
}

// -------- Phase 3: streaming log-likelihood reduction --------
__global__ void k_phase3(const float* __restrict__ pred,
                         const float* __restrict__ inp,
                         const int*   __restrict__ heart,
                         const float* __restrict__ mu, const float* __restrict__ h,
                         const float* __restrict__ Pk, float* __restrict__ num) {
    __shared__ float red[256];
    __shared__ float smu[12], sh[12], sP[4];
    const int b   = blockIdx.y;
    const int tid = threadIdx.x;
    if (tid < 12) { smu[tid] = mu[b*12 + tid]; sh[tid] = h[b*12 + tid]; }
    if (tid < 4)  { sP[tid]  = Pk[b*4 + tid]; }
    __syncthreads();

    const float* pb = pred  + (size_t)b*5*NPIX + NPIX;
    const float* ib = inp   + (size_t)b*3*NPIX;
    const int*   hb = heart + (size_t)b*NPIX;

    const int stride = gridDim.x * 256;
    float sum = 0.0f;
    for (int pix = blockIdx.x*256 + tid; pix < NPIX; pix += stride) {
        __builtin_prefetch(&pb[pix + stride], 0, 0);
        const float m  = (hb[pix] == 1) ? 1.0f : 0.0f;
        const float x0 = ib[0*NPIX + pix];
        const float x1 = ib[1*NPIX + pix];
        const float x2 = ib[2*NPIX + pix];
        float mix = 0.0f;
#pragma unroll
        for (int k = 0; k < 4; ++k) {
            const float p  = pb[k*NPIX + pix];
            const float d0 = x0 - smu[k*3+0];
            const float d1 = x1 - smu[k*3+1];
            const float d2 = x2 - smu[k*3+2];
            const float e  = sh[k*3+0]*d0*d0 + sh[k*3+1]*d1*d1 + sh[k*3+2]*d2*d2;
            mix += p*p*p * sP[k] * __expf(-e);
        }
        sum += __logf(mix + EPSV) * m;
    }

    red[tid] = sum;
    __syncthreads();
    for (int s = 128; s > 0; s >>= 1) {
        if (tid < s) red[tid] += red[tid + s];
        __syncthreads();
    }
    if (tid == 0) atomicAdd(&num[b], red[0]);
}

// -------- Final: scalar mean of -num[b]/nmask[b] --------
__global__ void k_final(const float* __restrict__ stats,
                        const float* __restrict__ num, float* __restrict__ out) {
    if (threadIdx.x == 0) {
        float s = 0.0f;
        for (int b = 0; b < BATCH; ++b) {
            const float nm = stats[b*40 + 32];   // row 4, col 0 = sum(mask)
            s += -num[b] / nm;
        }
        out[0] = s / (float)BATCH;
    }
}

extern "C" void kernel_launch(void* const* d_in, const int* in_sizes, int n_in,
                              void* d_out, int out_size, void* d_ws, size_t ws_size,
                              hipStream_t stream) {
    const float* pred  = (const float*)d_in[0];
    const float* inp   = (const float*)d_in[1];
    const int*   heart = (const int*)d_in[2];
    float* ws  = (float*)d_ws;
    float* out = (float*)d_out;

    k_zero<<<dim3((WS_TOTAL + 255)/256), 256, 0, stream>>>(ws);
    k_phase1<<<dim3(64, BATCH), 256, 0, stream>>>(pred, inp, heart, ws + WS_STATS);
    k_params<<<1, 64, 0, stream>>>(ws + WS_STATS, ws + WS_MU, ws + WS_H, ws + WS_PK);
    k_phase3<<<dim3(128, BATCH), 256, 0, stream>>>(pred, inp, heart,
                                                   ws + WS_MU, ws + WS_H, ws + WS_PK,
                                                   ws + WS_NUM);
    k_final<<<1, 32, 0, stream>>>(ws + WS_STATS, ws + WS_NUM, out);
}